// InteractionEncoder_12953621365121
// MI455X (gfx1250) — compile-verified
//
#include <hip/hip_runtime.h>
#include <hip/hip_bf16.h>
#include <math.h>

// ---------------------------------------------------------------------------
// InteractionEncoder for MI455X (gfx1250): all GEMMs on V_WMMA_F32_16X16X4_F32
// fp32 in / fp32 accumulate. Operand fragments are batch-preloaded into
// register arrays so the global/L2 loads pipeline 24-32 deep ahead of each
// WMMA chain instead of stalling per-step on s_wait_loadcnt 0.
// ---------------------------------------------------------------------------

typedef float v2f __attribute__((ext_vector_type(2)));
typedef float v8f __attribute__((ext_vector_type(8)));

#define DIM   128
#define NHEAD 6
#define HD    768
#define NAG   192
#define NLN   576
#define NND   768
#define LNEPS 1e-5f

// One K=4 step of D = A(16x4) * B(4x16) + C.
// Lane layout (wave32): lane L -> lm = L&15, hi = L>>4.
//   A: a.x = A[lm][kbase + 2*hi], a.y = A[lm][kbase + 2*hi + 1]
//   B: b.x = B[kbase + 2*hi][lm], b.y = B[kbase + 2*hi + 1][lm]
//   C: vgpr j -> C[j + 8*hi][lm]
__device__ __forceinline__ v8f wmma4v(v8f c, v2f a, v2f b) {
    return __builtin_amdgcn_wmma_f32_16x16x4_f32(false, a, false, b, (short)0, c,
                                                 false, false);
}

__device__ __forceinline__ v2f ld2(const float* p) { return *(const v2f*)p; }

__device__ __forceinline__ float node_val(const float* ag, const float* ln,
                                          int row, int col) {
    return (row < NAG) ? ag[row * DIM + col] : ln[(row - NAG) * DIM + col];
}
__device__ __forceinline__ v2f node_val2(const float* ag, const float* ln,
                                         int row, int col) {
    const float* p = (row < NAG) ? (ag + row * DIM + col)
                                 : (ln + (row - NAG) * DIM + col);
    return ld2(p);
}

// ---------------------------------------------------------------------------
// Kernel 1: q/k/v = LN(nodes @ W.T) (+ReLU for v).   grid = (48 row-tiles, 3)
// ---------------------------------------------------------------------------
__global__ __launch_bounds__(256) void k_qkv(
        const float* __restrict__ ag, const float* __restrict__ ln,
        const float* __restrict__ Wq, const float* __restrict__ gq, const float* __restrict__ bq,
        const float* __restrict__ Wk, const float* __restrict__ gk, const float* __restrict__ bk,
        const float* __restrict__ Wv, const float* __restrict__ gv, const float* __restrict__ bv,
        float* __restrict__ qb, float* __restrict__ kb, float* __restrict__ vb) {
    __shared__ float sA[16][128];      // input rows
    __shared__ float sAcc[16][772];    // 16 x 768 pre-LN result (padded)
    __shared__ float red[2][16][16];
    __shared__ float sMu[16], sRs[16];

    const int tid  = threadIdx.x;
    const int lane = tid & 31, wave = tid >> 5;
    const int lm = lane & 15, lk2 = (lane >> 4) * 2, half = lane >> 4;
    const int r0  = blockIdx.x * 16;
    const int sel = blockIdx.y;
    const float* W = sel == 0 ? Wq : (sel == 1 ? Wk : Wv);
    const float* g = sel == 0 ? gq : (sel == 1 ? gk : gv);
    const float* b = sel == 0 ? bq : (sel == 1 ? bk : bv);
    float* dst     = sel == 0 ? qb : (sel == 1 ? kb : vb);

    for (int e = tid; e < 16 * 128; e += 256) {
        int rr = e >> 7, cc = e & 127;
        sA[rr][cc] = node_val(ag, ln, r0 + rr, cc);
    }
    __syncthreads();

    // 48 col tiles of 16, 8 waves -> 6 tiles each. K = 128 (32 WMMA steps).
    for (int ct = wave; ct < 48; ct += 8) {
        const int n0 = ct * 16;
        const float* wrow = &W[(size_t)(n0 + lm) * DIM];   // B[k][n] = W[n][k]
        v2f af[32], bf[32];
        #pragma unroll
        for (int s = 0; s < 32; ++s) {
            const int kk = s * 4 + lk2;
            bf[s] = ld2(wrow + kk);          // 32 loads issued back-to-back
            af[s] = ld2(&sA[lm][kk]);
        }
        v8f c = {};
        #pragma unroll
        for (int s = 0; s < 32; ++s) c = wmma4v(c, af[s], bf[s]);
        #pragma unroll
        for (int j = 0; j < 8; ++j)
            sAcc[j + half * 8][n0 + lm] = c[j];
    }
    __syncthreads();

    // LayerNorm over HD=768 per row; 16 threads per row.
    const int tr = tid >> 4, tc = tid & 15;
    float s = 0.f, s2 = 0.f;
    for (int cdx = tc; cdx < HD; cdx += 16) {
        const float x = sAcc[tr][cdx];
        s += x; s2 += x * x;
    }
    red[0][tr][tc] = s; red[1][tr][tc] = s2;
    __syncthreads();
    if (tc == 0) {
        float a = 0.f, a2 = 0.f;
        #pragma unroll
        for (int i = 0; i < 16; ++i) { a += red[0][tr][i]; a2 += red[1][tr][i]; }
        const float mu  = a / (float)HD;
        const float var = a2 / (float)HD - mu * mu;
        sMu[tr] = mu; sRs[tr] = rsqrtf(var + LNEPS);
    }
    __syncthreads();
    const float mu = sMu[tr], rs = sRs[tr];
    float* drow = &dst[(size_t)(r0 + tr) * HD];
    for (int cdx = tc; cdx < HD; cdx += 16) {
        float y = (sAcc[tr][cdx] - mu) * rs * g[cdx] + b[cdx];
        if (sel == 2) y = fmaxf(y, 0.f);      // ReLU only on V
        drow[cdx] = y;
    }
}

// ---------------------------------------------------------------------------
// Kernel 2: per (graph, head) fused dense attention over 96 nodes.
// grid = (8 graphs, 6 heads), 256 threads (8 waves).
// ---------------------------------------------------------------------------
__global__ __launch_bounds__(256) void k_attn(
        const int* __restrict__ aid, const int* __restrict__ lid,
        const float* __restrict__ qb, const float* __restrict__ kb,
        const float* __restrict__ vb, float* __restrict__ attn) {
    __shared__ int   sIds[96];
    __shared__ float sS[96][100];   // scores -> probabilities
    __shared__ float sDen[96];

    const int tid  = threadIdx.x;
    const int lane = tid & 31, wave = tid >> 5;
    const int lm = lane & 15, lk2 = (lane >> 4) * 2, half = lane >> 4;
    const int bg = blockIdx.x, h = blockIdx.y;
    const float scale = 0.08838834764831845f;   // 128^-0.5

    if (tid < 96)
        sIds[tid] = (tid < 24) ? aid[bg * 24 + tid]
                               : NAG + lid[bg * 72 + (tid - 24)];
    __syncthreads();

    // S = Q K^T * scale : 6x6 tiles of 16x16, K = 128
    for (int t = wave; t < 36; t += 8) {
        const int ti = t / 6, tj = t % 6;
        const float* qrow = &qb[(size_t)sIds[ti * 16 + lm] * HD + h * DIM];
        const float* krow = &kb[(size_t)sIds[tj * 16 + lm] * HD + h * DIM];
        v2f af[32], bf[32];
        #pragma unroll
        for (int s = 0; s < 32; ++s) {
            const int kk = s * 4 + lk2;
            af[s] = ld2(qrow + kk);
            bf[s] = ld2(krow + kk);
        }
        v8f c = {};
        #pragma unroll
        for (int s = 0; s < 32; ++s) c = wmma4v(c, af[s], bf[s]);
        #pragma unroll
        for (int j = 0; j < 8; ++j)
            sS[ti * 16 + j + half * 8][tj * 16 + lm] = c[j] * scale;
    }
    __syncthreads();

    // Row softmax (row-max shift == global-max shift after normalization)
    if (tid < 96) {
        float mx = -3.402823466e38f;
        for (int j = 0; j < 96; ++j) mx = fmaxf(mx, sS[tid][j]);
        float sum = 0.f;
        for (int j = 0; j < 96; ++j) {
            const float p = expf(sS[tid][j] - mx);
            sS[tid][j] = p; sum += p;
        }
        sDen[tid] = sum;
    }
    __syncthreads();

    // O = P V : 6x8 tiles of 16x16, K = 96 (24 WMMA steps)
    for (int t = wave; t < 48; t += 8) {
        const int ti = t / 8, tn = t % 8;
        v2f af[24], bf[24];
        #pragma unroll
        for (int s = 0; s < 24; ++s) {
            const int kk = s * 4 + lk2;
            af[s]   = ld2(&sS[ti * 16 + lm][kk]);
            bf[s].x = vb[(size_t)sIds[kk]     * HD + h * DIM + tn * 16 + lm];
            bf[s].y = vb[(size_t)sIds[kk + 1] * HD + h * DIM + tn * 16 + lm];
        }
        v8f c = {};
        #pragma unroll
        for (int s = 0; s < 24; ++s) c = wmma4v(c, af[s], bf[s]);
        #pragma unroll
        for (int j = 0; j < 8; ++j) {
            const int row = ti * 16 + j + half * 8;
            attn[(size_t)sIds[row] * HD + h * DIM + tn * 16 + lm] = c[j] / sDen[row];
        }
    }
}

// shared-memory LayerNorm(+ReLU) helper over 128 cols, 16 rows, 256 threads
__device__ __forceinline__ void ln_relu_128(
        float (*src)[132], float (*dstS)[132],
        const float* __restrict__ g, const float* __restrict__ b,
        float red[2][16][16], float* sMu, float* sRs, int tid) {
    const int tr = tid >> 4, tc = tid & 15;
    float s = 0.f, s2 = 0.f;
    #pragma unroll
    for (int cdx = tc; cdx < DIM; cdx += 16) {
        const float x = src[tr][cdx];
        s += x; s2 += x * x;
    }
    red[0][tr][tc] = s; red[1][tr][tc] = s2;
    __syncthreads();
    if (tc == 0) {
        float a = 0.f, a2 = 0.f;
        #pragma unroll
        for (int i = 0; i < 16; ++i) { a += red[0][tr][i]; a2 += red[1][tr][i]; }
        const float mu  = a / (float)DIM;
        const float var = a2 / (float)DIM - mu * mu;
        sMu[tr] = mu; sRs[tr] = rsqrtf(var + LNEPS);
    }
    __syncthreads();
    const float mu = sMu[tr], rs = sRs[tr];
    #pragma unroll
    for (int cdx = tc; cdx < DIM; cdx += 16)
        dstS[tr][cdx] = fmaxf((src[tr][cdx] - mu) * rs * g[cdx] + b[cdx], 0.f);
}

// ---------------------------------------------------------------------------
// Kernel 3: fused tail — Wo1+LN+ReLU -> (Wo2 & nodes@W1.T accumulated) +LN+ReLU
//           -> W2 -> +residual ReLU.   grid = 48 row tiles, 256 threads.
// ---------------------------------------------------------------------------
__global__ __launch_bounds__(256) void k_tail(
        const float* __restrict__ ag, const float* __restrict__ ln,
        const float* __restrict__ attn,
        const float* __restrict__ Wo1, const float* __restrict__ go1, const float* __restrict__ bo1,
        const float* __restrict__ Wo2,
        const float* __restrict__ W1,  const float* __restrict__ gn,  const float* __restrict__ bn,
        const float* __restrict__ W2,  float* __restrict__ nf) {
    __shared__ float sR[16][132];
    __shared__ float sX[16][132];
    __shared__ float red[2][16][16];
    __shared__ float sMu[16], sRs[16];

    const int tid  = threadIdx.x;
    const int lane = tid & 31, wave = tid >> 5;
    const int lm = lane & 15, lk2 = (lane >> 4) * 2, half = lane >> 4;
    const int r0 = blockIdx.x * 16;
    const int n0 = wave * 16;            // each wave owns one 16-col tile of 128

    // GEMM1: attn[16x768] @ Wo1.T  (K = 768, processed in 128-wide K chunks)
    {
        const float* arow = &attn[(size_t)(r0 + lm) * HD];
        const float* brow = &Wo1[(size_t)(n0 + lm) * HD];
        v8f c = {};
        for (int ch = 0; ch < HD; ch += 128) {
            v2f af[32], bf[32];
            #pragma unroll
            for (int s = 0; s < 32; ++s) {
                const int kk = ch + s * 4 + lk2;
                af[s] = ld2(arow + kk);
                bf[s] = ld2(brow + kk);
            }
            #pragma unroll
            for (int s = 0; s < 32; ++s) c = wmma4v(c, af[s], bf[s]);
        }
        #pragma unroll
        for (int j = 0; j < 8; ++j) sR[j + half * 8][n0 + lm] = c[j];
    }
    __syncthreads();
    ln_relu_128(sR, sX, go1, bo1, red, sMu, sRs, tid);
    __syncthreads();

    // GEMM2 + GEMM3 share one accumulator: t1@Wo2.T + nodes@W1.T
    {
        const float* b2 = &Wo2[(size_t)(n0 + lm) * DIM];
        const float* b3 = &W1 [(size_t)(n0 + lm) * DIM];
        const int grow  = r0 + lm;
        v8f c = {};
        {
            v2f af[32], bf[32];
            #pragma unroll
            for (int s = 0; s < 32; ++s) {
                const int kk = s * 4 + lk2;
                af[s] = ld2(&sX[lm][kk]);
                bf[s] = ld2(b2 + kk);
            }
            #pragma unroll
            for (int s = 0; s < 32; ++s) c = wmma4v(c, af[s], bf[s]);
        }
        {
            v2f af[32], bf[32];
            #pragma unroll
            for (int s = 0; s < 32; ++s) {
                const int kk = s * 4 + lk2;
                af[s] = node_val2(ag, ln, grow, kk);
                bf[s] = ld2(b3 + kk);
            }
            #pragma unroll
            for (int s = 0; s < 32; ++s) c = wmma4v(c, af[s], bf[s]);
        }
        #pragma unroll
        for (int j = 0; j < 8; ++j) sR[j + half * 8][n0 + lm] = c[j];
    }
    __syncthreads();
    ln_relu_128(sR, sX, gn, bn, red, sMu, sRs, tid);
    __syncthreads();

    // GEMM4: h @ W2.T, then residual ReLU
    {
        const float* b4 = &W2[(size_t)(n0 + lm) * DIM];
        v2f af[32], bf[32];
        #pragma unroll
        for (int s = 0; s < 32; ++s) {
            const int kk = s * 4 + lk2;
            af[s] = ld2(&sX[lm][kk]);
            bf[s] = ld2(b4 + kk);
        }
        v8f c = {};
        #pragma unroll
        for (int s = 0; s < 32; ++s) c = wmma4v(c, af[s], bf[s]);
        #pragma unroll
        for (int j = 0; j < 8; ++j) {
            const int row = r0 + j + half * 8;
            const int col = n0 + lm;
            const float y = c[j] + node_val(ag, ln, row, col);
            nf[(size_t)row * DIM + col] = fmaxf(y, 0.f);
        }
    }
}

// ---------------------------------------------------------------------------
// Kernel 4: id-based gather into output (agents block, then lanes block)
// ---------------------------------------------------------------------------
__global__ __launch_bounds__(128) void k_scatter(
        const int* __restrict__ aid, const int* __restrict__ lid,
        const float* __restrict__ nf, float* __restrict__ out) {
    const int r = blockIdx.x, t = threadIdx.x;
    const int src = (r < NAG) ? aid[r] : NAG + lid[r - NAG];
    out[(size_t)r * DIM + t] = nf[(size_t)src * DIM + t];
}

// ---------------------------------------------------------------------------
extern "C" void kernel_launch(void* const* d_in, const int* in_sizes, int n_in,
                              void* d_out, int out_size, void* d_ws, size_t ws_size,
                              hipStream_t stream) {
    const float* agents    = (const float*)d_in[0];
    const int*   agent_ids = (const int*)  d_in[1];
    const float* lanes     = (const float*)d_in[2];
    const int*   lane_ids  = (const int*)  d_in[3];
    const float* Wq = (const float*)d_in[4];
    const float* gq = (const float*)d_in[5];
    const float* bq = (const float*)d_in[6];
    const float* Wk = (const float*)d_in[7];
    const float* gk = (const float*)d_in[8];
    const float* bk = (const float*)d_in[9];
    const float* Wv = (const float*)d_in[10];
    const float* gv = (const float*)d_in[11];
    const float* bv = (const float*)d_in[12];
    const float* Wo1 = (const float*)d_in[13];
    const float* go1 = (const float*)d_in[14];
    const float* bo1 = (const float*)d_in[15];
    const float* Wo2 = (const float*)d_in[16];
    const float* W1  = (const float*)d_in[17];
    const float* gn  = (const float*)d_in[18];
    const float* bn  = (const float*)d_in[19];
    const float* W2  = (const float*)d_in[20];

    float* qb   = (float*)d_ws;
    float* kb   = qb   + (size_t)NND * HD;
    float* vb   = kb   + (size_t)NND * HD;
    float* attn = vb   + (size_t)NND * HD;
    float* nf   = attn + (size_t)NND * HD;

    k_qkv<<<dim3(48, 3), 256, 0, stream>>>(agents, lanes,
                                           Wq, gq, bq, Wk, gk, bk, Wv, gv, bv,
                                           qb, kb, vb);
    k_attn<<<dim3(8, NHEAD), 256, 0, stream>>>(agent_ids, lane_ids, qb, kb, vb, attn);
    k_tail<<<48, 256, 0, stream>>>(agents, lanes, attn,
                                   Wo1, go1, bo1, Wo2, W1, gn, bn, W2, nf);
    k_scatter<<<NND, 128, 0, stream>>>(agent_ids, lane_ids, nf, (float*)d_out);
}